// SEGMeasure_27685359190287
// MI455X (gfx1250) — compile-verified
//
#include <hip/hip_runtime.h>

// SEG measure for 4096x4096 int32 label masks.
// Phases: zero ws -> LDS-privatized histogram (BW-bound: 134MB @ 23.3TB/s ~ 6us)
//         -> WMMA f32 16x16x4 row/col sums (padded, unguarded, vectorized loads)
//         -> single-WG sequential greedy scan.
//
// Confusion matrix stored PADDED: 208 rows (>=201, 13x16 blocks) x 272 cols
// (>=257, 17x16 blocks, 8B-aligned pairs). Padding is zero, so all WMMA loads
// are unguarded. Counts kept as f32 in global (exact for integers < 2^24).

#define NG 201                       // gt labels 0..200
#define NP 257                       // pred labels 0..256
#define NRP 208                      // padded rows   (13 * 16)
#define NPS 272                      // padded stride (17 * 16)
#define NBINS_P (NRP * NPS)          // 56576 words = 226,304 B LDS (< 320KB WGP)
#define GTS_OFF_W NBINS_P            // 64-word aligned (56576 % 64 == 0)
#define GTS_N 208
#define PRS_OFF_W (GTS_OFF_W + GTS_N)
#define PRS_N 272
#define TOTAL_W (PRS_OFF_W + PRS_N)

typedef float v2f __attribute__((ext_vector_type(2)));
typedef float v8f __attribute__((ext_vector_type(8)));

__global__ void seg_zero(unsigned* __restrict__ ws) {
  int i = blockIdx.x * blockDim.x + threadIdx.x;
  int stride = gridDim.x * blockDim.x;
  for (; i < TOTAL_W; i += stride) ws[i] = 0u;   // 0u == 0.0f bit pattern
}

// LDS-privatized confusion-matrix histogram. 226KB dynamic LDS per workgroup.
// int4 loads -> global_load_b128 for HBM bandwidth; ds atomics absorb the
// heavy diagonal contention (85% of pixels hit 201 bins); zero-skipping
// global_atomic_add_f32 flush.
__global__ void seg_hist(const int* __restrict__ pred, const int* __restrict__ gt,
                         float* __restrict__ counts, int n) {
  extern __shared__ unsigned lhist[];
  for (int i = threadIdx.x; i < NBINS_P; i += blockDim.x) lhist[i] = 0u;
  __syncthreads();

  int gid = blockIdx.x * blockDim.x + threadIdx.x;
  int stride = gridDim.x * blockDim.x;
  int n4 = n >> 2;
  const int4* p4 = (const int4*)pred;
  const int4* g4 = (const int4*)gt;
  for (int i = gid; i < n4; i += stride) {
    int4 p = p4[i];
    int4 g = g4[i];
    atomicAdd(&lhist[g.x * NPS + p.x], 1u);
    atomicAdd(&lhist[g.y * NPS + p.y], 1u);
    atomicAdd(&lhist[g.z * NPS + p.z], 1u);
    atomicAdd(&lhist[g.w * NPS + p.w], 1u);
  }
  for (int i = (n4 << 2) + gid; i < n; i += stride)
    atomicAdd(&lhist[gt[i] * NPS + pred[i]], 1u);
  __syncthreads();

  for (int i = threadIdx.x; i < NBINS_P; i += blockDim.x) {
    unsigned v = lhist[i];
    if (v) atomicAdd(&counts[i], (float)v);  // skip empty bins (most off-diag)
  }
}

// gt_sizes[row] = sum over pred columns of C[row][*], rows 0..200 (padded 208).
// One wave per 16-row block; D = A * ones(4x16) accumulated over K-chunks of 4.
// Unguarded aligned v2f (b64) loads; blockDim == 32 (wave32), EXEC all-ones.
__global__ void seg_rowsum_wmma(const float* __restrict__ cf,
                                float* __restrict__ gts) {
  int lane = threadIdx.x;            // 0..31
  int blk = blockIdx.x;              // 13 blocks of 16 rows
  int m = lane & 15;
  int row = blk * 16 + m;            // A-matrix: M = lane&15
  int kbase = (lane >> 4) * 2;       // halves own K pairs {0,1} / {2,3}
  const float* base = cf + row * NPS + kbase;
  v8f acc = {};
  v2f bones = {1.0f, 1.0f};          // all-ones B: K attribution irrelevant
  for (int k0 = 0; k0 < NPS; k0 += 4) {
    v2f a = *(const v2f*)(base + k0);     // 8B-aligned (row*272+even offset)
    acc = __builtin_amdgcn_wmma_f32_16x16x4_f32(false, a, false, bones,
                                                (short)0, acc, false, false);
  }
  // Row sums at N=0: lane 0 -> rows blk*16+0..7 in acc[0..7], lane 16 -> +8..15.
  if (m == 0) {
    int rbase = blk * 16 + (lane >> 4) * 8;
    for (int r = 0; r < 8; ++r) gts[rbase + r] = acc[r];
  }
}

// pred_sizes[col] = sum over gt rows of C[*][col], cols 0..256 (padded 272).
// D = ones(16x4) * B accumulated over K-chunks of 4 rows; unguarded loads.
__global__ void seg_colsum_wmma(const float* __restrict__ cf,
                                float* __restrict__ prs) {
  int lane = threadIdx.x;
  int blk = blockIdx.x;              // 17 blocks of 16 cols
  int col = blk * 16 + (lane & 15);
  int kbase = (lane >> 4) * 2;
  const float* base = cf + kbase * NPS + col;
  v8f acc = {};
  v2f aones = {1.0f, 1.0f};
  for (int k0 = 0; k0 < 204; k0 += 4) {    // rows 0..203 cover NG=201, padded 0
    v2f b;
    b.x = base[k0 * NPS];
    b.y = base[(k0 + 1) * NPS];
    acc = __builtin_amdgcn_wmma_f32_16x16x4_f32(false, aones, false, b,
                                                (short)0, acc, false, false);
  }
  // Column sums in D row 0: lanes 0..15, acc[0] = sum of column (blk*16+lane).
  if (lane < 16) prs[col] = acc[0];
}

// Sequential greedy matching: 200 steps, max/argmax (first-index tie-break,
// matching jnp.argmax) over 256 pred columns each step. One workgroup.
__global__ void seg_finalize(const float* __restrict__ cf,
                             const float* __restrict__ gts,
                             const float* __restrict__ prs,
                             float* __restrict__ out) {
  __shared__ float s_val[256];
  __shared__ int s_idx[256];
  __shared__ int s_avail[256];
  int t = threadIdx.x;               // pred column index 0..255 (label t+1)
  float psize = prs[t + 1];          // pred_sizes over C[:,1:]
  s_avail[t] = (psize > 0.0f) ? 1 : 0;
  __syncthreads();

  for (int i = 0; i < NG - 1; ++i) {       // gt labels 1..200
    int g = i + 1;
    float gsize = gts[g];
    float inter = cf[g * NPS + (t + 1)];
    float jval = 0.0f;
    if (s_avail[t] && (inter > 0.5f * gsize)) {
      float denom = fmaxf((gsize + psize) - inter, 1.0f);
      jval = inter / denom;
    }
    s_val[t] = jval;
    s_idx[t] = t;
    __syncthreads();
    for (int off = 128; off > 0; off >>= 1) {
      if (t < off) {
        float v2 = s_val[t + off];
        int i2 = s_idx[t + off];
        if (v2 > s_val[t] || (v2 == s_val[t] && i2 < s_idx[t])) {
          s_val[t] = v2;
          s_idx[t] = i2;
        }
      }
      __syncthreads();
    }
    if (t == 0) {
      float best = s_val[0];
      out[1 + i] = best;                   // seg entry for gt label i+1
      if (best > 0.0f) s_avail[s_idx[0]] = 0;
    }
    __syncthreads();
  }

  if (t == 0) {
    int npresent = 0;
    float ssum = 0.0f;
    for (int i = 0; i < NG - 1; ++i) {
      if (gts[i + 1] > 0.0f) ++npresent;
      ssum += out[1 + i];                  // seg already 0 where gt absent
    }
    if (npresent < 1) npresent = 1;
    out[0] = ssum / (float)npresent;       // seg_avg
  }
}

extern "C" void kernel_launch(void* const* d_in, const int* in_sizes, int n_in,
                              void* d_out, int out_size, void* d_ws, size_t ws_size,
                              hipStream_t stream) {
  const int* pred = (const int*)d_in[0];   // pred_labels_mask
  const int* gt = (const int*)d_in[1];     // gt_labels_mask
  int n = in_sizes[0];                     // 4096*4096
  unsigned* ws = (unsigned*)d_ws;
  float* counts = (float*)ws;              // padded 208 x 272 f32 matrix
  float* gts = (float*)(ws + GTS_OFF_W);
  float* prs = (float*)(ws + PRS_OFF_W);
  float* out = (float*)d_out;

  seg_zero<<<64, 256, 0, stream>>>(ws);
  seg_hist<<<128, 256, NBINS_P * sizeof(unsigned), stream>>>(pred, gt, counts, n);
  seg_rowsum_wmma<<<13, 32, 0, stream>>>(counts, gts);
  seg_colsum_wmma<<<17, 32, 0, stream>>>(counts, prs);
  seg_finalize<<<1, 256, 0, stream>>>(counts, gts, prs, out);
}